// BoundaryDoULoss_11020886082202
// MI455X (gfx1250) — compile-verified
//
#include <hip/hip_runtime.h>
#include <math.h>

typedef __attribute__((ext_vector_type(2))) float v2f;
typedef __attribute__((ext_vector_type(8))) float v8f;

#define NBATCH 16
#define NCLS   4
#define HH     512
#define WW     512
#define HWSZ   (HH * WW)                 // 262144
#define NPIX   (NBATCH * HWSZ)           // 4,194,304
#define NBLK   1024
#define NTHR   256
#define NWAVE  (NTHR / 32)
#define SMOOTH 1e-5f

// ---------------------------------------------------------------------------
// Stage 1: fused softmax + boundary detection + per-class partial reductions.
// One 16-float partial row per block -> part[NBLK][16], layout j = q*4 + c:
//   q=0: intersect (sum p*onehot), q=1: z_sum (sum p^2),
//   q=2: S (region count),         q=3: C (boundary count)
// ---------------------------------------------------------------------------
__global__ __launch_bounds__(NTHR) void bdl_stage1(
    const float* __restrict__ logits,   // [16,4,512,512]
    const int*   __restrict__ tgt,      // [16,512,512]
    float*       __restrict__ part)     // [NBLK,16]
{
    float accI0 = 0.f, accI1 = 0.f, accI2 = 0.f, accI3 = 0.f;
    float accZ0 = 0.f, accZ1 = 0.f, accZ2 = 0.f, accZ3 = 0.f;
    float accS0 = 0.f, accS1 = 0.f, accS2 = 0.f, accS3 = 0.f;
    float accB0 = 0.f, accB1 = 0.f, accB2 = 0.f, accB3 = 0.f;

    const int tid    = threadIdx.x;
    const int stride = NBLK * NTHR;     // 262144 threads -> 16 pixels each

    for (int idx = blockIdx.x * NTHR + tid; idx < NPIX; idx += stride) {
        const int b  = idx / HWSZ;
        const int hw = idx - b * HWSZ;
        const int h  = hw >> 9;
        const int w  = hw & (WW - 1);

        const int t  = tgt[idx];
        // OOB neighbor counts as "different class" (zero padding breaks Y==5)
        const int up = (h > 0)      ? tgt[idx - WW] : -1;
        const int dn = (h < HH - 1) ? tgt[idx + WW] : -1;
        const int lf = (w > 0)      ? tgt[idx - 1]  : -1;
        const int rt = (w < WW - 1) ? tgt[idx + 1]  : -1;
        const float bnd = ((up != t) | (dn != t) | (lf != t) | (rt != t)) ? 1.0f : 0.0f;

        // softmax over the 4 channel logits (channel stride = HWSZ floats)
        const size_t base = (size_t)b * (NCLS * HWSZ) + (size_t)hw;
        const float x0 = __builtin_nontemporal_load(logits + base);
        const float x1 = __builtin_nontemporal_load(logits + base + (size_t)HWSZ);
        const float x2 = __builtin_nontemporal_load(logits + base + (size_t)(2 * HWSZ));
        const float x3 = __builtin_nontemporal_load(logits + base + (size_t)(3 * HWSZ));

        const float m  = fmaxf(fmaxf(x0, x1), fmaxf(x2, x3));
        const float e0 = expf(x0 - m);
        const float e1 = expf(x1 - m);
        const float e2 = expf(x2 - m);
        const float e3 = expf(x3 - m);
        const float inv = 1.0f / (e0 + e1 + e2 + e3);
        const float p0 = e0 * inv, p1 = e1 * inv, p2 = e2 * inv, p3 = e3 * inv;

        accZ0 += p0 * p0;  accZ1 += p1 * p1;  accZ2 += p2 * p2;  accZ3 += p3 * p3;

        const float q0 = (t == 0) ? 1.0f : 0.0f;
        const float q1 = (t == 1) ? 1.0f : 0.0f;
        const float q2 = (t == 2) ? 1.0f : 0.0f;
        const float q3 = (t == 3) ? 1.0f : 0.0f;

        accI0 += p0 * q0;  accI1 += p1 * q1;  accI2 += p2 * q2;  accI3 += p3 * q3;
        accS0 += q0;       accS1 += q1;       accS2 += q2;       accS3 += q3;
        accB0 += q0 * bnd; accB1 += q1 * bnd; accB2 += q2 * bnd; accB3 += q3 * bnd;
    }

    float acc[16] = { accI0, accI1, accI2, accI3,
                      accZ0, accZ1, accZ2, accZ3,
                      accS0, accS1, accS2, accS3,
                      accB0, accB1, accB2, accB3 };

    __shared__ float red[NWAVE * 16];
    const int lane = tid & 31;
    const int wave = tid >> 5;

    #pragma unroll
    for (int j = 0; j < 16; ++j) {
        float v = acc[j];
        v += __shfl_down(v, 16, 32);
        v += __shfl_down(v, 8, 32);
        v += __shfl_down(v, 4, 32);
        v += __shfl_down(v, 2, 32);
        v += __shfl_down(v, 1, 32);
        if (lane == 0) red[wave * 16 + j] = v;
    }
    __syncthreads();

    if (tid < 16) {
        float s = 0.f;
        #pragma unroll
        for (int wv = 0; wv < NWAVE; ++wv) s += red[wv * 16 + tid];
        part[blockIdx.x * 16 + tid] = s;
    }
}

// ---------------------------------------------------------------------------
// Stage 2: one wave32. Column-sum the [NBLK,16] partial matrix with
// V_WMMA_F32_16X16X4_F32 using A = ones(16x4):
//   D[m,n] = sum_k B[k,n] + C[m,n]  -> invariant to K<->lane permutation.
// C/D layout: VGPR0, lanes 0..15 hold (M=0, N=lane) -> lane j has total_j.
// ---------------------------------------------------------------------------
__global__ __launch_bounds__(32) void bdl_stage2(
    const float* __restrict__ part,     // [NBLK,16]
    float*       __restrict__ out)      // [1]
{
    const int lane = threadIdx.x;
    v8f c = {};
    v2f a = {1.0f, 1.0f};               // all-ones A (16x4), any lane mapping

    const int col  = lane & 15;          // N = lane % 16 (B/C/D striping)
    const int rsel = (lane >> 4) << 1;   // lanes 0-15: rows r+0,r+1; 16-31: r+2,r+3

    for (int r = 0; r < NBLK; r += 4) {
        v2f bv;
        bv[0] = part[(r + rsel)     * 16 + col];
        bv[1] = part[(r + rsel + 1) * 16 + col];
        c = __builtin_amdgcn_wmma_f32_16x16x4_f32(
                /*neg_a=*/false, a, /*neg_b=*/false, bv,
                /*c_mod=*/(short)0, c, /*reuse_a=*/false, /*reuse_b=*/false);
    }

    __shared__ float T[16];
    if (lane < 16) T[lane] = c[0];       // M=0 row: the 16 column totals
    __syncthreads();

    if (lane == 0) {
        float loss = 0.0f;
        #pragma unroll
        for (int cc = 0; cc < 4; ++cc) {
            const float I  = T[cc];
            const float Z  = T[4 + cc];
            const float S  = T[8 + cc];   // also equals y_sum
            const float Cb = T[12 + cc];
            float alpha = 1.0f - (Cb + SMOOTH) / (S + SMOOTH);
            alpha = fminf(2.0f * alpha - 1.0f, 0.8f);
            const float num = Z + S - 2.0f * I + SMOOTH;
            const float den = Z + S - (1.0f + alpha) * I + SMOOTH;
            loss += num / den;
        }
        out[0] = loss * 0.25f;
    }
}

extern "C" void kernel_launch(void* const* d_in, const int* in_sizes, int n_in,
                              void* d_out, int out_size, void* d_ws, size_t ws_size,
                              hipStream_t stream) {
    const float* logits = (const float*)d_in[0];
    const int*   tgt    = (const int*)d_in[1];
    float*       out    = (float*)d_out;
    float*       part   = (float*)d_ws;          // needs NBLK*16*4 = 64 KB

    bdl_stage1<<<NBLK, NTHR, 0, stream>>>(logits, tgt, part);
    bdl_stage2<<<1, 32, 0, stream>>>(part, out);
}